// MambaPredictor_90580860273032
// MI455X (gfx1250) — compile-verified
//
#include <hip/hip_runtime.h>
#include <math.h>

#define BATCH  8
#define SEQ    1024
#define DMODEL 512
#define DSTATE 16
#define DCONV  4
#define DINNER 1024
#define DTRANK 32

// __builtin_amdgcn_sched_group_barrier masks
#define SG_MFMA_WMMA 0x0008
#define SG_VMEM_READ 0x0020

typedef __attribute__((ext_vector_type(2))) float v2f;
typedef __attribute__((ext_vector_type(8))) float v8f;

__device__ __forceinline__ float sigmoidf_(float x) { return 1.0f / (1.0f + __expf(-x)); }
__device__ __forceinline__ float siluf_(float x)    { return x * sigmoidf_(x); }
__device__ __forceinline__ float softplusf_(float x){ return (x > 20.0f) ? x : log1pf(__expf(x)); }

// ---------------------------------------------------------------------------
// Generic fp32 WMMA GEMM:  C[m,n] = act( sum_k A[m,k]*B[n,k] + bias[n] )
// A: M x K row-major (lda), B: N x K row-major (ldb), C: M x N row-major (ldc)
//
// One wave computes a 16x64 output strip (4 accumulators) with
// V_WMMA_F32_16X16X4_F32 over K in steps of 4. Software-pipelined via
// register double-buffering + sched_group_barrier directives: each iteration
// issues the NEXT step's 5 VMEM reads, then the CURRENT step's 4 WMMAs, so
// the wait before the WMMA burst covers loads that are a full iteration old.
//
// No lane predication in the hot loop: partial M-tiles clamp the A row index
// (redundant loads of row M-1; garbage accumulator rows are never stored).
//
// fp32 A-tile layout: lanes 0-15 hold rows m0+lane with K=(k,k+1); lanes
// 16-31 hold the same rows with K=(k+2,k+3) -> one aligned float2 per lane.
// C/D layout: VGPR v -> row m0 + 8*(lane>=16) + v, col n0 + (lane&15).
//
// Requirements honored by all launches below: N % 64 == 0, K % 4 == 0,
// K >= 8, lda/ldb even (float2-aligned rows).
// act: 0 = none, 1 = softplus
// ---------------------------------------------------------------------------
__global__ void wmma_gemm_nt(const float* __restrict__ A, int lda,
                             const float* __restrict__ B, int ldb,
                             float* __restrict__ C, int ldc,
                             int M, int N, int K,
                             const float* __restrict__ bias, int act)
{
    const int lane = threadIdx.x & 31;
    const int wave = blockIdx.x * (blockDim.x >> 5) + (threadIdx.x >> 5);
    const int tmN = (M + 15) >> 4;      // 16-row tiles
    const int tnN = N >> 6;             // 64-col strips
    if (wave >= tmN * tnN) return;      // wave-uniform: EXEC stays full
    const int tm = wave / tnN;
    const int tn = wave - tm * tnN;
    const int m0 = tm << 4, n0 = tn << 6;

    const int half = lane >> 4;         // 0: K pair (k,k+1); 1: (k+2,k+3)
    const int l16  = lane & 15;

    int am = m0 + l16;                  // A row this lane feeds
    if (am >= M) am = M - 1;            // clamp: no predication in hot loop
    const int bn = n0 + l16;            // base B row (= output column) of strip

    const float* arow  = A + (size_t)am * lda + 2 * half;
    const float* brow0 = B + (size_t)(bn +  0) * ldb + 2 * half;
    const float* brow1 = B + (size_t)(bn + 16) * ldb + 2 * half;
    const float* brow2 = B + (size_t)(bn + 32) * ldb + 2 * half;
    const float* brow3 = B + (size_t)(bn + 48) * ldb + 2 * half;

    v8f acc0 = {}, acc1 = {}, acc2 = {}, acc3 = {};

    // prologue: fragments for k = 0
    v2f a  = *(const v2f*)(arow);
    v2f b0 = *(const v2f*)(brow0);
    v2f b1 = *(const v2f*)(brow1);
    v2f b2 = *(const v2f*)(brow2);
    v2f b3 = *(const v2f*)(brow3);

    // steady state: issue k+4 loads, then k's WMMAs (pipelined by the
    // sched_group_barrier directives below)
    for (int k = 0; k + 4 < K; k += 4) {
        const v2f an  = *(const v2f*)(arow  + k + 4);
        const v2f bn0 = *(const v2f*)(brow0 + k + 4);
        const v2f bn1 = *(const v2f*)(brow1 + k + 4);
        const v2f bn2 = *(const v2f*)(brow2 + k + 4);
        const v2f bn3 = *(const v2f*)(brow3 + k + 4);
        acc0 = __builtin_amdgcn_wmma_f32_16x16x4_f32(false, a, false, b0, (short)0, acc0, false, false);
        acc1 = __builtin_amdgcn_wmma_f32_16x16x4_f32(false, a, false, b1, (short)0, acc1, false, false);
        acc2 = __builtin_amdgcn_wmma_f32_16x16x4_f32(false, a, false, b2, (short)0, acc2, false, false);
        acc3 = __builtin_amdgcn_wmma_f32_16x16x4_f32(false, a, false, b3, (short)0, acc3, false, false);
        a = an; b0 = bn0; b1 = bn1; b2 = bn2; b3 = bn3;
        // pipeline directive: 5 VMEM reads first, then 4 WMMAs per iteration
        __builtin_amdgcn_sched_group_barrier(SG_VMEM_READ, 5, 0);
        __builtin_amdgcn_sched_group_barrier(SG_MFMA_WMMA, 4, 0);
    }
    // epilogue: last K-step
    acc0 = __builtin_amdgcn_wmma_f32_16x16x4_f32(false, a, false, b0, (short)0, acc0, false, false);
    acc1 = __builtin_amdgcn_wmma_f32_16x16x4_f32(false, a, false, b1, (short)0, acc1, false, false);
    acc2 = __builtin_amdgcn_wmma_f32_16x16x4_f32(false, a, false, b2, (short)0, acc2, false, false);
    acc3 = __builtin_amdgcn_wmma_f32_16x16x4_f32(false, a, false, b3, (short)0, acc3, false, false);

    const int rbase = m0 + half * 8;
    v8f accs[4] = {acc0, acc1, acc2, acc3};
#pragma unroll
    for (int j = 0; j < 4; ++j) {
        const int col = n0 + 16 * j + l16;
        const float bs = bias ? bias[col] : 0.0f;
#pragma unroll
        for (int v = 0; v < 8; ++v) {
            const int r = rbase + v;
            if (r < M) {
                float val = accs[j][v] + bs;
                if (act == 1) val = softplusf_(val);
                C[(size_t)r * ldc + col] = val;
            }
        }
    }
}

// ---------------------------------------------------------------------------
// Depthwise causal conv(4) + SiLU over xi = xz[..., :DINNER]
// ---------------------------------------------------------------------------
__global__ void conv_silu_k(const float* __restrict__ xz,
                            const float* __restrict__ cw,
                            const float* __restrict__ cb,
                            float* __restrict__ xc)
{
    const int idx = blockIdx.x * blockDim.x + threadIdx.x;
    if (idx >= BATCH * SEQ * DINNER) return;
    const int d  = idx & (DINNER - 1);
    const int bt = idx >> 10;                      // b*SEQ + t
    const int t  = bt & (SEQ - 1);
    float acc = cb[d];
#pragma unroll
    for (int k = 0; k < DCONV; ++k) {
        const int tt = t - (DCONV - 1) + k;
        if (tt >= 0)
            acc += cw[d * DCONV + k] *
                   xz[(size_t)(bt + tt - t) * (2 * DINNER) + d];
    }
    xc[idx] = siluf_(acc);
}

// ---------------------------------------------------------------------------
// Selective scan: one thread per (b,d); 16 states in registers; B_t staged in
// LDS per 128-step chunk; only y at t = SEQ-1 is materialized (that is all the
// reference consumes). Epilogue fuses +D*xc and *silu(z).
// ---------------------------------------------------------------------------
#define SCAN_CH 128
__global__ void scan_k(const float* __restrict__ dt,
                       const float* __restrict__ xc,
                       const float* __restrict__ dbc,   // (B*SEQ, 64): [dt_rank | B | C]
                       const float* __restrict__ xz,    // for z = xz[..., DINNER:]
                       const float* __restrict__ A_log,
                       const float* __restrict__ Dp,
                       float* __restrict__ ylast)       // (BATCH, DINNER)
{
    const int b = blockIdx.x >> 2;
    const int d = ((blockIdx.x & 3) << 8) + threadIdx.x;

    __shared__ float sB[SCAN_CH * DSTATE];

    float Av[DSTATE], h[DSTATE];
#pragma unroll
    for (int s = 0; s < DSTATE; ++s) {
        Av[s] = -__expf(A_log[d * DSTATE + s]);
        h[s] = 0.0f;
    }

    const size_t base = (size_t)b * SEQ;
    for (int t0 = 0; t0 < SEQ; t0 += SCAN_CH) {
        // stage B_t chunk (shared across all d in this batch) into LDS
        for (int i = threadIdx.x; i < SCAN_CH * DSTATE; i += blockDim.x) {
            const int tt = i >> 4;
            const int s  = i & 15;
            sB[i] = dbc[(base + t0 + tt) * 64 + DTRANK + s];
        }
        __syncthreads();

        if (t0 + SCAN_CH < SEQ) {   // warm next chunk's dt/xc lines (global_prefetch_b8)
            __builtin_prefetch(&dt[(base + t0 + SCAN_CH) * DINNER + d], 0, 0);
            __builtin_prefetch(&xc[(base + t0 + SCAN_CH) * DINNER + d], 0, 0);
        }

        for (int tt = 0; tt < SCAN_CH; ++tt) {
            const size_t ti = (base + t0 + tt) * DINNER + d;
            const float dtv = dt[ti];
            const float du  = dtv * xc[ti];
            const float* bs = &sB[tt * DSTATE];
#pragma unroll
            for (int s = 0; s < DSTATE; ++s)
                h[s] = __expf(dtv * Av[s]) * h[s] + du * bs[s];
        }
        __syncthreads();
    }

    float y = 0.0f;
#pragma unroll
    for (int s = 0; s < DSTATE; ++s)
        y += h[s] * dbc[(base + SEQ - 1) * 64 + DTRANK + DSTATE + s];

    const float xcl = xc[(base + SEQ - 1) * DINNER + d];
    const float zl  = xz[(base + SEQ - 1) * (2 * DINNER) + DINNER + d];
    ylast[b * DINNER + d] = (y + Dp[d] * xcl) * siluf_(zl);
}

// ---------------------------------------------------------------------------
// Row LayerNorm over DMODEL elements (one block per batch row)
// ---------------------------------------------------------------------------
__global__ void layernorm_k(const float* __restrict__ in,
                            const float* __restrict__ g,
                            const float* __restrict__ beta,
                            float* __restrict__ out, int N)
{
    const int b = blockIdx.x;
    const float* row = in + (size_t)b * N;
    __shared__ float red[256];

    float s = 0.0f;
    for (int i = threadIdx.x; i < N; i += blockDim.x) s += row[i];
    red[threadIdx.x] = s;
    __syncthreads();
    for (int off = 128; off > 0; off >>= 1) {
        if ((int)threadIdx.x < off) red[threadIdx.x] += red[threadIdx.x + off];
        __syncthreads();
    }
    const float mu = red[0] / (float)N;
    __syncthreads();

    float v = 0.0f;
    for (int i = threadIdx.x; i < N; i += blockDim.x) {
        const float dx = row[i] - mu;
        v += dx * dx;
    }
    red[threadIdx.x] = v;
    __syncthreads();
    for (int off = 128; off > 0; off >>= 1) {
        if ((int)threadIdx.x < off) red[threadIdx.x] += red[threadIdx.x + off];
        __syncthreads();
    }
    const float rstd = rsqrtf(red[0] / (float)N + 1e-5f);
    for (int i = threadIdx.x; i < N; i += blockDim.x)
        out[(size_t)b * N + i] = (row[i] - mu) * rstd * g[i] + beta[i];
}

// ---------------------------------------------------------------------------
static inline int gemm_blocks(int M, int N)
{
    const long long waves = (long long)((M + 15) / 16) * (N / 64);
    return (int)((waves + 7) / 8);     // 256 threads = 8 waves per block
}

extern "C" void kernel_launch(void* const* d_in, const int* in_sizes, int n_in,
                              void* d_out, int out_size, void* d_ws, size_t ws_size,
                              hipStream_t stream)
{
    const float* x      = (const float*)d_in[0];
    const float* W_in   = (const float*)d_in[1];
    const float* conv_w = (const float*)d_in[2];
    const float* conv_b = (const float*)d_in[3];
    const float* W_x    = (const float*)d_in[4];
    const float* W_dt   = (const float*)d_in[5];
    const float* b_dt   = (const float*)d_in[6];
    const float* A_log  = (const float*)d_in[7];
    const float* Dp     = (const float*)d_in[8];
    const float* W_out  = (const float*)d_in[9];
    const float* ln_g   = (const float*)d_in[10];
    const float* ln_b   = (const float*)d_in[11];
    const float* head_W = (const float*)d_in[12];
    const float* head_b = (const float*)d_in[13];

    float* ws      = (float*)d_ws;
    float* xz      = ws;                                         // 8*1024*2048
    float* xc      = xz  + (size_t)BATCH * SEQ * 2 * DINNER;     // 8*1024*1024
    float* dbc     = xc  + (size_t)BATCH * SEQ * DINNER;         // 8*1024*64
    float* dtb     = dbc + (size_t)BATCH * SEQ * 64;             // 8*1024*1024
    float* ylast   = dtb + (size_t)BATCH * SEQ * DINNER;         // 8*1024
    float* outlast = ylast   + BATCH * DINNER;                   // 8*512
    float* lastn   = outlast + BATCH * DMODEL;                   // 8*512

    const int M = BATCH * SEQ;

    // 1) xz = x @ W_in^T            (8192 x 2048, K=512)
    wmma_gemm_nt<<<gemm_blocks(M, 2 * DINNER), 256, 0, stream>>>(
        x, DMODEL, W_in, DMODEL, xz, 2 * DINNER, M, 2 * DINNER, DMODEL, nullptr, 0);

    // 2) xc = silu(depthwise_conv(xi) + conv_b)
    conv_silu_k<<<(BATCH * SEQ * DINNER) / 256, 256, 0, stream>>>(
        xz, conv_w, conv_b, xc);

    // 3) dbc = xc @ W_x^T           (8192 x 64, K=1024)
    wmma_gemm_nt<<<gemm_blocks(M, 64), 256, 0, stream>>>(
        xc, DINNER, W_x, DINNER, dbc, 64, M, 64, DINNER, nullptr, 0);

    // 4) dt = softplus(dbc[:, :32] @ W_dt^T + b_dt)   (8192 x 1024, K=32)
    wmma_gemm_nt<<<gemm_blocks(M, DINNER), 256, 0, stream>>>(
        dbc, 64, W_dt, DTRANK, dtb, DINNER, M, DINNER, DTRANK, b_dt, 1);

    // 5) selective scan -> fused gate -> ylast (only t = SEQ-1 is consumed)
    scan_k<<<BATCH * 4, 256, 0, stream>>>(dtb, xc, dbc, xz, A_log, Dp, ylast);

    // 6) outlast = ylast @ W_out^T  (8 x 512, K=1024)
    wmma_gemm_nt<<<gemm_blocks(BATCH, DMODEL), 256, 0, stream>>>(
        ylast, DINNER, W_out, DINNER, outlast, DMODEL, BATCH, DMODEL, DINNER, nullptr, 0);

    // 7) layernorm
    layernorm_k<<<BATCH, 256, 0, stream>>>(outlast, ln_g, ln_b, lastn, DMODEL);

    // 8) result = lastn @ head_W^T + head_b  (8 x 512, K=512) -> d_out
    wmma_gemm_nt<<<gemm_blocks(BATCH, DMODEL), 256, 0, stream>>>(
        lastn, DMODEL, head_W, DMODEL, (float*)d_out, DMODEL, BATCH, DMODEL, DMODEL,
        head_b, 0);
}